// IndependenceLoss_label_65034394796691
// MI455X (gfx1250) — compile-verified
//
#include <hip/hip_runtime.h>
#include <cstdint>

#define NCLS   32
#define NCOND  6
#define NBATCH 4
#define KGRP   (NCOND * NBATCH)      // 24 groups
#define CCH    32                    // C (matches wave32 exactly)
#define LSTACK 4                     // L
#define EPSF   1e-10f
#define JCOLS  (KGRP * NCLS * CCH)   // 24576 joint columns per l
#define CNTCOL (KGRP * NCLS)         // 768 count bins
#define PMAX   128                   // max partial blocks per l

typedef __attribute__((ext_vector_type(2))) float v2f;
typedef __attribute__((ext_vector_type(8))) float v8f;

__device__ __forceinline__ float wave_sum(float v) {
    #pragma unroll
    for (int m = 16; m > 0; m >>= 1) v += __shfl_xor(v, m, 32);
    return v;
}

// ---------------- Phase 1: streaming joint histogram (LDS ds_add_f32) -------
__global__ void joint_hist_kernel(const float* __restrict__ probs,
                                  const int*   __restrict__ assign,
                                  const int*   __restrict__ cond,
                                  const int*   __restrict__ batch,
                                  float*       __restrict__ jpart,
                                  int N, int P)
{
    extern __shared__ float lds[];               // JCOLS floats = 96 KB
    const int tid  = threadIdx.x;
    const int nthr = blockDim.x;
    const int lane = tid & 31;
    const int wid  = tid >> 5;
    const int nw   = nthr >> 5;
    const int l    = blockIdx.x / P;
    const int p    = blockIdx.x % P;

    for (int i = tid; i < JCOLS; i += nthr) lds[i] = 0.0f;
    __syncthreads();

    const long long chunk = ((long long)N + P - 1) / P;
    const long long n0 = (long long)p * chunk;
    long long n1 = n0 + chunk; if (n1 > N) n1 = N;
    const float* pl = probs + (size_t)l * (size_t)N * CCH;

    for (long long base = n0 + (long long)wid * 32; base < n1;
         base += (long long)nw * 32) {
        long long idx  = base + lane;
        long long cidx = idx < n1 ? idx : (n1 - 1);
        int a   = assign[cidx];
        int g   = cond[cidx] * NBATCH + batch[cidx];
        int seg = g * NCLS + a;                  // [0, 768)
        int kmax = (int)((n1 - base < 32) ? (n1 - base) : 32);   // uniform
        for (int k = 0; k < kmax; ++k) {
            int segk = __builtin_amdgcn_readlane(seg, k);        // SGPR broadcast
            float v = __builtin_nontemporal_load(&pl[(size_t)(base + k) * CCH + lane]);
            atomicAdd(&lds[segk * CCH + lane], v);               // ds_add_f32, no bank conflicts
        }
    }
    __syncthreads();
    float* dst = jpart + ((size_t)l * P + p) * JCOLS;
    for (int i = tid; i < JCOLS; i += nthr) dst[i] = lds[i];
}

// ---------------- Phase 1b: assignment counts --------------------------------
__global__ void counts_kernel(const int* __restrict__ assign,
                              const int* __restrict__ cond,
                              const int* __restrict__ batch,
                              float* __restrict__ cpart, int N, int P)
{
    __shared__ float scnt[CNTCOL];
    const int tid = threadIdx.x;
    const int p   = blockIdx.x;
    for (int i = tid; i < CNTCOL; i += blockDim.x) scnt[i] = 0.0f;
    __syncthreads();
    const long long chunk = ((long long)N + P - 1) / P;
    const long long n0 = (long long)p * chunk;
    long long n1 = n0 + chunk; if (n1 > N) n1 = N;
    for (long long n = n0 + tid; n < n1; n += blockDim.x) {
        int seg = (cond[n] * NBATCH + batch[n]) * NCLS + assign[n];
        atomicAdd(&scnt[seg], 1.0f);
    }
    __syncthreads();
    float* dst = cpart + (size_t)p * CNTCOL;
    for (int i = tid; i < CNTCOL; i += blockDim.x) dst[i] = scnt[i];
}

// ---------------- Phase 2: WMMA ones-GEMM reduction of partials --------------
// D = ones(16x4) x B(4x16)  accumulated over P/4 chunks; D row 0 = column sums.
__global__ void wmma_reduce_kernel(const float* __restrict__ jpart,
                                   const float* __restrict__ cpart,
                                   float* __restrict__ joint,
                                   float* __restrict__ counts, int P)
{
    const int lane = threadIdx.x & 31;
    const int wid  = threadIdx.x >> 5;
    const int wpb  = blockDim.x >> 5;
    const int jt_tiles = LSTACK * (JCOLS / 16);        // 6144
    const int total    = jt_tiles + (CNTCOL / 16);     // +48

    v2f a; a.x = 1.0f; a.y = 1.0f;                     // ones A-matrix

    for (int t = blockIdx.x * wpb + wid; t < total; t += gridDim.x * wpb) {
        const float* src; float* out; int stride, m0;
        if (t < jt_tiles) {
            int l  = t / (JCOLS / 16);
            m0     = (t % (JCOLS / 16)) * 16;
            src    = jpart + (size_t)l * P * JCOLS;
            stride = JCOLS;
            out    = joint + (size_t)l * JCOLS + m0;
        } else {
            m0     = (t - jt_tiles) * 16;
            src    = cpart;
            stride = CNTCOL;
            out    = counts + m0;
        }
        const int col = m0 + (lane & 15);
        const int rb  = (lane < 16) ? 0 : 2;           // B: 4x16 f32 in 2 VGPRs
        v8f c = {0.f, 0.f, 0.f, 0.f, 0.f, 0.f, 0.f, 0.f};
        for (int kc = 0; kc < P; kc += 4) {
            v2f b;
            b.x = src[(size_t)(kc + rb)     * stride + col];
            b.y = src[(size_t)(kc + rb + 1) * stride + col];
            c = __builtin_amdgcn_wmma_f32_16x16x4_f32(
                    false, a, false, b, (short)0, c, false, false);
        }
        if (lane < 16) out[col] = c[0];                // D row 0: N = lane
    }
}

// ---------------- Phase 3: entropies + MI -> scalar --------------------------
__global__ void finalize_kernel(const float* __restrict__ joint,
                                const float* __restrict__ counts,
                                float* __restrict__ out)
{
    __shared__ float mi_arr[LSTACK * KGRP];
    const int lane = threadIdx.x & 31;
    const int wid  = threadIdx.x >> 5;
    const int nw   = blockDim.x >> 5;

    for (int pair = wid; pair < LSTACK * KGRP; pair += nw) {
        int l = pair / KGRP, g = pair % KGRP;
        const float* jb = joint + (size_t)l * JCOLS + g * (NCLS * CCH);

        float jt = 0.0f;
        for (int tpos = lane; tpos < NCLS * CCH; tpos += 32) jt += jb[tpos];
        jt = wave_sum(jt);
        float dj = fmaxf(jt, EPSF);

        float hab = 0.0f;
        for (int tpos = lane; tpos < NCLS * CCH; tpos += 32) {
            float pv = jb[tpos] / dj;
            hab -= pv * logf(pv + EPSF);
        }
        hab = wave_sum(hab);

        float ps = 0.0f;                               // probsum[:, lane] = sum_i joint
        for (int i = 0; i < NCLS; ++i) ps += jb[i * CCH + lane];
        float pb = ps / dj;
        float hb = wave_sum(-pb * logf(pb + EPSF));

        float ca = counts[g * NCLS + lane];
        float ng = wave_sum(ca);
        float pa = ca / fmaxf(ng, 1.0f);
        float ha = wave_sum(-pa * logf(pa + EPSF));

        float denom = ha + hb;
        float mi = 2.0f * (denom - hab) / fmaxf(denom, EPSF);
        if (!(ng > 0.0f)) mi = 0.0f;
        if (lane == 0) mi_arr[pair] = mi;
    }
    __syncthreads();
    if (threadIdx.x == 0) {                            // deterministic final sum
        int np = 0;
        for (int g = 0; g < KGRP; ++g) {
            float ng = 0.0f;
            for (int i = 0; i < NCLS; ++i) ng += counts[g * NCLS + i];
            if (ng > 0.0f) ++np;
        }
        float acc = 0.0f;
        for (int pr = 0; pr < LSTACK * KGRP; ++pr) acc += mi_arr[pr];
        acc /= (float)LSTACK;                          // mean over l
        out[0] = acc / fmaxf((float)np, 1.0f);         // SCALER == 1
    }
}

// ---------------- host-side launcher -----------------------------------------
extern "C" void kernel_launch(void* const* d_in, const int* in_sizes, int n_in,
                              void* d_out, int out_size, void* d_ws, size_t ws_size,
                              hipStream_t stream) {
    const int*   assign = (const int*)  d_in[0];
    const float* probs  = (const float*)d_in[1];
    const int*   cond   = (const int*)  d_in[2];
    const int*   batch  = (const int*)  d_in[3];
    const int N = in_sizes[0];

    // Choose partial count P (multiple of 4) that fits the workspace.
    // floats needed = (P + 1) * (LSTACK*JCOLS + CNTCOL)
    const size_t perP = (size_t)LSTACK * JCOLS + CNTCOL;       // 99072 floats
    long long pfit = (long long)(ws_size / sizeof(float) / perP) - 1;
    int P = (int)((pfit > PMAX) ? PMAX : (pfit < 4 ? 4 : pfit));
    P &= ~3;

    float* ws     = (float*)d_ws;
    float* jpart  = ws;                                        // L*P*JCOLS
    float* cpart  = jpart + (size_t)LSTACK * P * JCOLS;        // P*CNTCOL
    float* joint  = cpart + (size_t)P * CNTCOL;                // L*JCOLS
    float* counts = joint + (size_t)LSTACK * JCOLS;            // CNTCOL

    joint_hist_kernel<<<LSTACK * P, 256, JCOLS * sizeof(float), stream>>>(
        probs, assign, cond, batch, jpart, N, P);
    counts_kernel<<<P, 256, 0, stream>>>(assign, cond, batch, cpart, N, P);
    wmma_reduce_kernel<<<256, 256, 0, stream>>>(jpart, cpart, joint, counts, P);
    finalize_kernel<<<1, 256, 0, stream>>>(joint, counts, (float*)d_out);
}